// SupConLoss_4612794875952
// MI455X (gfx1250) — compile-verified
//
#include <hip/hip_runtime.h>
#include <hip/hip_bf16.h>
#include <math.h>
#include <stdint.h>

typedef __attribute__((ext_vector_type(2))) float v2f;
typedef __attribute__((ext_vector_type(4))) float v4f;
typedef __attribute__((ext_vector_type(8))) float v8f;

#define KDIM 128
#define TEMP_INV (1.0f / 0.07f)
#define EPS_F 1e-8f

// B tile in LDS: 16 rows x 128 floats, padded to 132-float stride for
// conflict-free ds_load_b64 (lanes 0-15 hit banks =0,1 mod 4; lanes 16-31
// hit banks =2,3 mod 4).
#define BROW 132
#define BTILE (16 * BROW)   // floats per buffer

// ---------------------------------------------------------------------------
// Kernel 1: L2-normalize rows (one wave per 128-elem row) + labels -> int32.
// ---------------------------------------------------------------------------
__global__ void supcon_norm_kernel(const float* __restrict__ feat,
                                   const long long* __restrict__ lab64,
                                   float* __restrict__ fn,
                                   int* __restrict__ lab32) {
    const int wave = blockIdx.x * (blockDim.x >> 5) + (threadIdx.x >> 5);
    const int lane = threadIdx.x & 31;
    const float* src = feat + (size_t)wave * KDIM + lane * 4;
    v4f x = *(const v4f*)src;
    float ss = x.x * x.x + x.y * x.y + x.z * x.z + x.w * x.w;
#pragma unroll
    for (int off = 16; off >= 1; off >>= 1)
        ss += __shfl_xor(ss, off, 32);
    const float nrm = fmaxf(sqrtf(ss), 1e-12f);
    const float inv = 1.0f / nrm;
    v4f y = {x.x * inv, x.y * inv, x.z * inv, x.w * inv};
    *(v4f*)(fn + (size_t)wave * KDIM + lane * 4) = y;
    if (lane == 0) lab32[wave] = (int)lab64[wave];
}

// ---------------------------------------------------------------------------
// Cooperative async prefetch of one 16x128 column tile into LDS (padded).
// 512 b128 chunks, 256 threads -> 2 per thread.  Tracked on ASYNCcnt.
// ---------------------------------------------------------------------------
__device__ __forceinline__ void prefetch_tile(const float* __restrict__ fn,
                                              int colBase, uint32_t ldsByteBase,
                                              int tid) {
#pragma unroll
    for (int s = 0; s < 2; ++s) {
        const int linear = s * 256 + tid;       // 0..511
        const int r  = linear >> 5;             // tile row 0..15
        const int ch = linear & 31;             // 16B chunk 0..31
        const float* g = fn + (size_t)(colBase + r) * KDIM + ch * 4;
        const uint32_t loff = ldsByteBase + (uint32_t)((r * BROW + ch * 4) * 4);
        asm volatile("global_load_async_to_lds_b128 %0, %1, off"
                     :: "v"(loff), "v"(g)
                     : "memory");
    }
}

// ---------------------------------------------------------------------------
// Kernel 2: fused sim / softmax-denominator / positive-mask statistics.
// One wave owns 16 rows (A block pinned in 64 VGPRs); column tiles stream
// through double-buffered LDS via async-to-LDS copies shared by all 8 waves.
// 16x16 sim tile = 32x V_WMMA_F32_16X16X4_F32 (K=128, exact fp32).
// Constant shift m = 1/T (the self-sim row max); shift-invariance of
// log-softmax makes this identical to the reference's per-row max subtract.
// ---------------------------------------------------------------------------
__global__ void __launch_bounds__(256, 1)
supcon_main_kernel(const float* __restrict__ fn,
                   const int* __restrict__ lab32,
                   float* __restrict__ mlp,
                   float* __restrict__ validf,
                   int N) {
    __shared__ float ldsB[2 * BTILE];

    const int tid  = threadIdx.x;
    const int wave = blockIdx.x * (blockDim.x >> 5) + (tid >> 5);
    const int lane = tid & 31;
    const int idx  = lane & 15;   // N-col within tile / M-row within A tile
    const int h    = lane >> 4;   // half -> K sub-pair selector
    const int rowBase = wave * 16;

    const uint32_t ldsBase = (uint32_t)(uintptr_t)(&ldsB[0]);

    // Pin A block: lane holds row (rowBase+idx), K elems {4t+2h, 4t+2h+1}.
    const float* arow = fn + (size_t)(rowBase + idx) * KDIM + 2 * h;
    v2f areg[32];
#pragma unroll
    for (int t = 0; t < 32; ++t)
        areg[t] = *(const v2f*)(arow + 4 * t);

    int labRow[8];
#pragma unroll
    for (int v = 0; v < 8; ++v)
        labRow[v] = lab32[rowBase + v + 8 * h];

    float S[8], P[8];
    int   C[8];
#pragma unroll
    for (int v = 0; v < 8; ++v) { S[v] = 0.0f; P[v] = 0.0f; C[v] = 0; }

    const int numTiles = N >> 4;

    // Prime buffer 0 with tile 0.
    prefetch_tile(fn, 0, ldsBase, tid);

    for (int jt = 0; jt < numTiles; ++jt) {
        // Overlap: issue next tile's async copies, then drain current tile.
        if (jt + 1 < numTiles) {
            prefetch_tile(fn, (jt + 1) * 16,
                          ldsBase + (uint32_t)(((jt + 1) & 1) * BTILE * 4), tid);
            asm volatile("s_wait_asynccnt 0x2" ::: "memory");
        } else {
            asm volatile("s_wait_asynccnt 0x0" ::: "memory");
        }
        __syncthreads();   // current tile visible to all waves

        const int colBase = jt * 16;
        const int labCol = lab32[colBase + idx];
        const float* bptr = ldsB + (jt & 1) * BTILE + idx * BROW + 2 * h;

        v8f c = {};
#pragma unroll
        for (int t = 0; t < 32; ++t) {
            v2f b = *(const v2f*)(bptr + 4 * t);  // ds_load_b64, bank-clean
            c = __builtin_amdgcn_wmma_f32_16x16x4_f32(
                    false, areg[t], false, b, (short)0, c, false, false);
        }

        const int gcol = colBase + idx;
#pragma unroll
        for (int v = 0; v < 8; ++v) {
            const int grow = rowBase + v + 8 * h;
            const float sim = c[v] * TEMP_INV;
            const bool isSelf = (grow == gcol);
            const bool isPos  = (labRow[v] == labCol) && !isSelf;
            const float e = __expf(sim - TEMP_INV);   // shift by row max (=1/T)
            S[v] += isSelf ? 0.0f : e;
            P[v] += isPos ? sim : 0.0f;
            C[v] += isPos ? 1 : 0;
        }
        __syncthreads();   // everyone done reading before buffer is re-written
    }

    // Reduce across the 16 lanes of each half (xor masks 1..8 stay in-half).
#pragma unroll
    for (int v = 0; v < 8; ++v) {
        float ss = S[v], pp = P[v], cc = (float)C[v];
#pragma unroll
        for (int off = 1; off < 16; off <<= 1) {
            ss += __shfl_xor(ss, off, 32);
            pp += __shfl_xor(pp, off, 32);
            cc += __shfl_xor(cc, off, 32);
        }
        if (idx == 0) {
            const int grow = rowBase + v + 8 * h;
            // mean_log_prob = (P - C*max - C*log(S+eps)) / (C+eps), max = 1/T
            const float logd = __logf(ss + EPS_F);
            const float mlpv = (pp - cc * (TEMP_INV + logd)) / (cc + EPS_F);
            const bool valid = cc > 0.0f;
            mlp[grow]    = valid ? mlpv : 0.0f;
            validf[grow] = valid ? 1.0f : 0.0f;
        }
    }
}

// ---------------------------------------------------------------------------
// Kernel 3: deterministic final reduction to the scalar loss.
// ---------------------------------------------------------------------------
__global__ void supcon_reduce_kernel(const float* __restrict__ mlp,
                                     const float* __restrict__ validf,
                                     float* __restrict__ out, int N) {
    __shared__ float ws_s[32];
    __shared__ float ws_c[32];
    const int tid = threadIdx.x;
    float s = 0.0f, c = 0.0f;
    for (int i = tid; i < N; i += blockDim.x) {
        s += mlp[i];
        c += validf[i];
    }
#pragma unroll
    for (int off = 16; off >= 1; off >>= 1) {
        s += __shfl_xor(s, off, 32);
        c += __shfl_xor(c, off, 32);
    }
    const int wid = tid >> 5, lane = tid & 31;
    if (lane == 0) { ws_s[wid] = s; ws_c[wid] = c; }
    __syncthreads();
    if (wid == 0) {
        const int nw = blockDim.x >> 5;
        s = (lane < nw) ? ws_s[lane] : 0.0f;
        c = (lane < nw) ? ws_c[lane] : 0.0f;
#pragma unroll
        for (int off = 16; off >= 1; off >>= 1) {
            s += __shfl_xor(s, off, 32);
            c += __shfl_xor(c, off, 32);
        }
        if (lane == 0)
            out[0] = (c > 0.0f) ? (-s / fmaxf(c, 1.0f)) : 0.0f;
    }
}

// ---------------------------------------------------------------------------
extern "C" void kernel_launch(void* const* d_in, const int* in_sizes, int n_in,
                              void* d_out, int out_size, void* d_ws, size_t ws_size,
                              hipStream_t stream) {
    const float* feat      = (const float*)d_in[0];
    const long long* lab64 = (const long long*)d_in[1];
    float* out             = (float*)d_out;

    const int N = in_sizes[0] / KDIM;   // 8192

    // Workspace layout
    float* fn     = (float*)d_ws;                        // N*KDIM floats
    int*   lab32  = (int*)(fn + (size_t)N * KDIM);       // N ints
    float* mlp    = (float*)(lab32 + N);                 // N floats
    float* validf = mlp + N;                             // N floats

    supcon_norm_kernel<<<N / 8, 256, 0, stream>>>(feat, lab64, fn, lab32);
    supcon_main_kernel<<<(N / 16) / 8, 256, 0, stream>>>(fn, lab32, mlp, validf, N);
    supcon_reduce_kernel<<<1, 1024, 0, stream>>>(mlp, validf, out, N);
}